// HartleySpectralConv_747324310072
// MI455X (gfx1250) — compile-verified
//
#include <hip/hip_runtime.h>
#include <hip/hip_bf16.h>
#include <math.h>

// ---------------------------------------------------------------------------
// HartleySpectralConv on MI455X (gfx1250, wave32).
//   x: (16,32,256,256) f32   weight: (32,32,64,64) f32   bias: (32) f32
//   out: (16,32,256,256) f32
// Mode-cropped DHT == dense GEMMs vs cos/sin bases on V_WMMA_F32_16X16X4_F32.
// x tiles staged LDS via Tensor Data Mover (double-buffered, TENSORcnt).
// ---------------------------------------------------------------------------

typedef __attribute__((ext_vector_type(2))) float v2f;
typedef __attribute__((ext_vector_type(8))) float v8f;
typedef __attribute__((ext_vector_type(4))) unsigned int v4u;
typedef __attribute__((ext_vector_type(8))) int v8i;
typedef __attribute__((ext_vector_type(4))) int v4i;

__device__ __forceinline__ v8f wmma_f32(v2f a, v2f b, v8f c) {
  return __builtin_amdgcn_wmma_f32_16x16x4_f32(false, a, false, b, (short)0, c,
                                               false, false);
}

#if __has_builtin(__builtin_amdgcn_tensor_load_to_lds)
#define USE_TDM 1
#else
#define USE_TDM 0
#endif

#if USE_TDM
// 2D TDM load: tile rows x 256 f32, global row stride 256 elems, into LDS at
// lds_off with +4-dword pad every 256 dwords (-> 260-float LDS row stride).
__device__ __forceinline__ void tdm_load_x(unsigned lds_off, const float* gsrc,
                                           int rows) {
  unsigned long long ga = (unsigned long long)(uintptr_t)gsrc;
  v4u g0;
  g0.x = 1u;                                     // count=1, is_restore=0
  g0.y = lds_off;                                // lds_addr
  g0.z = (unsigned)(ga & 0xFFFFFFFFu);           // global_addr[31:0]
  g0.w = (unsigned)((ga >> 32) & 0x01FFFFFFu) | 0x80000000u;  // [56:32]|type=2
  v8i g1;
  // data_size=2(4B) | pad_enable | pad_interval=7(256dw) | pad_amount=3(4dw)
  g1[0] = 0x07D20000;
  g1[1] = 256 << 16;                             // tensor_dim0 = 256
  g1[2] = (rows & 0xFFFF) << 16;                 // tensor_dim1 = rows
  g1[3] = 256 << 16;                             // tile_dim0 = 256
  g1[4] = rows & 0xFFFF;                         // tile_dim1 = rows
  g1[5] = 256;                                   // tensor_dim0_stride lo32
  g1[6] = 0;
  g1[7] = 0;
  v4i z4 = {};
  v8i z8 = {};
  // 6-arg toolchain form: (g0, g1, g2, g3, g4, cpol)
  __builtin_amdgcn_tensor_load_to_lds(g0, g1, z4, z4, z8, 0);
}
#endif

// ---------------------------------------------------------------------------
// Basis tables:
//  Bw  [256][128]: col c<64: cos(2pi*(96+c)*h/256), c>=64: sin     (A of inv)
//  Bu  [192][256]: rows 0..63 cos, 64..127 sin, 128..191 -sin      (GEMM1 B /
//                  GEMM2 A; rows r<128 are exactly Bw transposed)
//  CasT[512][64] : row r<256: (cos+sin)(f_v*r)/65536 ; r>=256: (cos-sin)/65536
// ---------------------------------------------------------------------------
__global__ void tables_kernel(float* __restrict__ Bw, float* __restrict__ Bu,
                              float* __restrict__ CasT) {
  int idx = blockIdx.x * 256 + threadIdx.x;
  if (idx < 256 * 128) {
    int h = idx >> 7, c = idx & 127;
    int f = 96 + (c & 63);
    float t = (float)((f * h) & 255) * (1.0f / 128.0f);
    Bw[idx] = (c < 64) ? cospif(t) : sinpif(t);
  }
  if (idx < 192 * 256) {
    int r = idx >> 8, h = idx & 255;
    int f = 96 + (r & 63);
    float t = (float)((f * h) & 255) * (1.0f / 128.0f);
    float cs = cospif(t), sn = sinpif(t);
    Bu[idx] = (r < 64) ? cs : ((r < 128) ? sn : -sn);
  }
  if (idx < 512 * 64) {
    int r = idx >> 6, v = idx & 63;
    int w = r & 255;
    int f = 96 + v;
    float t = (float)((f * w) & 255) * (1.0f / 128.0f);
    float cs = cospif(t), sn = sinpif(t);
    float cas = (r < 256) ? (cs + sn) : (cs - sn);
    CasT[idx] = cas * (1.0f / 65536.0f);
  }
}

// ---------------------------------------------------------------------------
// Weight transpose: WT[mode][term][i*32+o]
//   term 0: W[i,o,u,v]   term 1: W[i,o,(64-u)%64,(64-v)%64]
// ---------------------------------------------------------------------------
__global__ void wtrans_kernel(const float* __restrict__ W,
                              float* __restrict__ WT) {
  int idx = blockIdx.x * 256 + threadIdx.x;   // 4096*2*1024
  int io = idx & 1023;
  int o = io & 31, i = io >> 5;
  int term = (idx >> 10) & 1;
  int mode = idx >> 11;
  int u = mode >> 6, v = mode & 63;
  int uu = term ? ((64 - u) & 63) : u;
  int vv = term ? ((64 - v) & 63) : v;
  WT[idx] = W[(((size_t)i * 32 + o) * 64 + uu) * 64 + vv];
}

// ---------------------------------------------------------------------------
// Stage 1+2 fused, per image (b,i), 8 blocks of 32 h-rows, TDM double-buffer:
//   Yr/Yi[h,v] = sum_w x[h,w]*basis  -> yblkT[v'][h] (transposed, LDS)
//   E = Cu*Yr - Su*Yi ; O = Su*Yr + Cu*Yi  (VGPR accumulators)
// EOm layout (mode-major): EOm[u*64+v][isO*512 + i*16 + b]
// ---------------------------------------------------------------------------
__global__ __launch_bounds__(256) void fwd_kernel(const float* __restrict__ x,
                                                  const float* __restrict__ Bu,
                                                  float* __restrict__ EOm) {
  __shared__ __align__(16) float xbuf[2][32][260];
  __shared__ __align__(16) float yblkT[128][36];   // [v'][h], padded

  const int img  = blockIdx.x;            // b*32 + i
  const int bb   = img >> 5, ii = img & 31;
  const int lane = threadIdx.x & 31;
  const int wave = threadIdx.x >> 5;
  const int lm   = lane & 15;
  const int kb   = (lane >> 4) << 1;
  const int mh   = (lane >> 4) << 3;
  const float* xi = x + (size_t)img * 65536;

  v8f acc[4];
  #pragma unroll
  for (int t = 0; t < 4; ++t) acc[t] = {};

#if USE_TDM
  const unsigned xbase = (unsigned)(uintptr_t)&xbuf[0][0][0];
  if (wave == 0) tdm_load_x(xbase, xi, 32);
#endif

  for (int hb = 0; hb < 8; ++hb) {
    const int p = hb & 1;
#if USE_TDM
    if (wave == 0) {
      if (hb < 7) {
        tdm_load_x(xbase + ((hb + 1) & 1) * (32 * 260 * 4),
                   xi + (size_t)(hb + 1) * 8192, 32);
        __builtin_amdgcn_s_wait_tensorcnt(1);   // current buffer ready
      } else {
        __builtin_amdgcn_s_wait_tensorcnt(0);
      }
    }
    __syncthreads();
#else
    __syncthreads();
    for (int idx = threadIdx.x; idx < 2048; idx += 256) {
      int r = idx >> 6, c4 = idx & 63;
      *(float4*)&xbuf[p][r][c4 * 4] =
          ((const float4*)(xi + (size_t)(hb * 32 + r) * 256))[c4];
    }
    __syncthreads();
#endif

    // GEMM1: yblkT(128 x 32) = xbuf(32x256) x Bu[0:128](B, n-major); 2 tiles/wave
    for (int t = 0; t < 2; ++t) {
      int tid = wave * 2 + t;
      int tm = tid >> 3, tn = tid & 7;
      v8f c = {};
      const float* arow = &xbuf[p][tm * 16 + lm][0];
      const float* brow = Bu + (size_t)(tn * 16 + lm) * 256;
      #pragma unroll 4
      for (int k0 = 0; k0 < 256; k0 += 4) {
        v2f a = {arow[k0 + kb], arow[k0 + kb + 1]};
        v2f b = {brow[k0 + kb], brow[k0 + kb + 1]};
        c = wmma_f32(a, b, c);
      }
      int n = tn * 16 + lm;
      int mb = tm * 16 + mh;
      float4 lo = {c[0], c[1], c[2], c[3]};
      float4 hi = {c[4], c[5], c[6], c[7]};
      *(float4*)&yblkT[n][mb]     = lo;
      *(float4*)&yblkT[n][mb + 4] = hi;
    }
    __syncthreads();

    // GEMM2 accumulate: 32 tiles (0..15 E, 16..31 O), 4/wave, K=32 this block
    for (int t = 0; t < 4; ++t) {
      int tid = wave * 4 + t;
      bool isO = tid >= 16;
      int te = tid & 15;
      int ut = te >> 2, vt = te & 3;
      int rA1 = isO ? 64 : 0;    // O: Su*Yr   E: Cu*Yr
      int rA2 = isO ? 0 : 128;   // O: Cu*Yi   E: -Su*Yi
      const float* a1 = Bu + (size_t)(rA1 + ut * 16 + lm) * 256 + hb * 32;
      const float* a2 = Bu + (size_t)(rA2 + ut * 16 + lm) * 256 + hb * 32;
      const float* b1 = &yblkT[vt * 16 + lm][0];        // Yr (v' = v)
      const float* b2 = &yblkT[64 + vt * 16 + lm][0];   // Yi (v' = 64+v)
      v8f c = acc[t];
      #pragma unroll 4
      for (int k0 = 0; k0 < 32; k0 += 4) {
        v2f a = {a1[k0 + kb], a1[k0 + kb + 1]};
        v2f b = {b1[k0 + kb], b1[k0 + kb + 1]};
        c = wmma_f32(a, b, c);
      }
      #pragma unroll 4
      for (int k0 = 0; k0 < 32; k0 += 4) {
        v2f a = {a2[k0 + kb], a2[k0 + kb + 1]};
        v2f b = {b2[k0 + kb], b2[k0 + kb + 1]};
        c = wmma_f32(a, b, c);
      }
      acc[t] = c;
    }
  }

  // store EO tiles, mode-major
  for (int t = 0; t < 4; ++t) {
    int tid = wave * 4 + t;
    int isO = (tid >= 16) ? 512 : 0;
    int te = tid & 15;
    int ut = te >> 2, vt = te & 3;
    int nv = vt * 16 + lm;
    int mb = ut * 16 + mh;
    v8f c = acc[t];
    #pragma unroll
    for (int r = 0; r < 8; ++r)
      EOm[(size_t)((mb + r) * 64 + nv) * 1024 + isO + ii * 16 + bb] = c[r];
  }
}

// ---------------------------------------------------------------------------
// Stage 3: per mode: S[b,o] = sum_i E[b,i]*W + O[b,i]*Wrev  (coalesced loads)
// SPEC layout: [(b*32+o)][mode]
// ---------------------------------------------------------------------------
__global__ __launch_bounds__(64) void mix_kernel(const float* __restrict__ EOm,
                                                 const float* __restrict__ WT,
                                                 float* __restrict__ SPEC) {
  const int mode = blockIdx.x;
  const int lane = threadIdx.x & 31;
  const int wave = threadIdx.x >> 5;
  const int lm = lane & 15;
  const int kb = (lane >> 4) << 1;
  const int mh = (lane >> 4) << 3;
  const int o = wave * 16 + lm;

  const float* Eb = EOm + (size_t)mode * 1024 + lm;      // + i*16  (E) / +512 (O)
  const float* Wb = WT + (size_t)mode * 2048 + o;        // + i*32  (term0) / +1024

  v8f c = {};
  #pragma unroll
  for (int k0 = 0; k0 < 32; k0 += 4) {
    int i0 = k0 + kb;
    v2f a = {Eb[i0 * 16], Eb[i0 * 16 + 16]};
    v2f b = {Wb[i0 * 32], Wb[i0 * 32 + 32]};
    c = wmma_f32(a, b, c);
  }
  #pragma unroll
  for (int k0 = 0; k0 < 32; k0 += 4) {
    int i0 = k0 + kb;
    v2f a = {Eb[512 + i0 * 16], Eb[512 + i0 * 16 + 16]};
    v2f b = {Wb[1024 + i0 * 32], Wb[1024 + i0 * 32 + 32]};
    c = wmma_f32(a, b, c);
  }
  #pragma unroll
  for (int r = 0; r < 8; ++r)
    SPEC[(size_t)((mh + r) * 32 + o) * 4096 + mode] = c[r];
}

// ---------------------------------------------------------------------------
// Stage 4 fused, per (b,o) image:
//   TbT[w][u'] = spec(64x64) x CasT (T1 at u'<64, T2 at u'>=64)
//   out(256x256) = Bw(256x128) x [T1;T2]  (+bias)
// ---------------------------------------------------------------------------
__global__ __launch_bounds__(256) void inv_kernel(const float* __restrict__ SPEC,
                                                  const float* __restrict__ CasT,
                                                  const float* __restrict__ Bw,
                                                  const float* __restrict__ bias,
                                                  float* __restrict__ out) {
  __shared__ __align__(16) float spec[64][68];
  __shared__ __align__(16) float TbT[256][132];   // [w][u'], padded

  const int img  = blockIdx.x;  // b*32 + o
  const int o    = img & 31;
  const int lane = threadIdx.x & 31;
  const int wave = threadIdx.x >> 5;
  const int lm   = lane & 15;
  const int kb   = (lane >> 4) << 1;
  const int mh   = (lane >> 4) << 3;

  for (int idx = threadIdx.x; idx < 1024; idx += 256) {
    int r = idx >> 4, c4 = idx & 15;
    *(float4*)&spec[r][c4 * 4] = ((const float4*)(SPEC + (size_t)img * 4096))[idx];
  }
  __syncthreads();

  // GEMM 4a: 128 tiles (4x32), 16/wave, K=64
  for (int t = 0; t < 16; ++t) {
    int tid = wave * 16 + t;
    int tm = tid >> 5, tn = tid & 31;
    v8f c = {};
    const float* arow = &spec[tm * 16 + lm][0];
    const float* brow = CasT + (size_t)(tn * 16 + lm) * 64;
    #pragma unroll 4
    for (int k0 = 0; k0 < 64; k0 += 4) {
      v2f a = {arow[k0 + kb], arow[k0 + kb + 1]};
      v2f b = {brow[k0 + kb], brow[k0 + kb + 1]};
      c = wmma_f32(a, b, c);
    }
    int n = tn * 16 + lm;
    int w = n & 255;
    int ub = tm * 16 + mh + ((n < 256) ? 0 : 64);
    float4 lo = {c[0], c[1], c[2], c[3]};
    float4 hi = {c[4], c[5], c[6], c[7]};
    *(float4*)&TbT[w][ub]     = lo;
    *(float4*)&TbT[w][ub + 4] = hi;
  }
  __syncthreads();

  // GEMM 4b: 256 tiles (16x16), 32/wave, K=128
  const float bo = bias[o];
  float* oimg = out + (size_t)img * 65536;
  for (int t = 0; t < 32; ++t) {
    int tid = wave * 32 + t;
    int tm = tid >> 4, tn = tid & 15;
    v8f c = {};
    const float* arow = Bw + (size_t)(tm * 16 + lm) * 128;
    const float* brow = &TbT[tn * 16 + lm][0];
    #pragma unroll 4
    for (int k0 = 0; k0 < 128; k0 += 4) {
      v2f a = {arow[k0 + kb], arow[k0 + kb + 1]};
      v2f b = {brow[k0 + kb], brow[k0 + kb + 1]};
      c = wmma_f32(a, b, c);
    }
    int n = tn * 16 + lm;
    int mb = tm * 16 + mh;
    #pragma unroll
    for (int r = 0; r < 8; ++r)
      oimg[(size_t)(mb + r) * 256 + n] = c[r] + bo;
  }
}

// ---------------------------------------------------------------------------
extern "C" void kernel_launch(void* const* d_in, const int* in_sizes, int n_in,
                              void* d_out, int out_size, void* d_ws, size_t ws_size,
                              hipStream_t stream) {
  const float* x    = (const float*)d_in[0];
  const float* W    = (const float*)d_in[1];
  const float* bias = (const float*)d_in[2];
  float* out = (float*)d_out;

  float* ws   = (float*)d_ws;
  float* Bw   = ws;                  //    32,768 floats
  float* Bu   = ws + 32768;          //    49,152
  float* CasT = ws + 81920;          //    32,768
  float* EOm  = ws + 114688;         // 4,194,304  (4096 x 1024)
  float* SPEC = ws + 4308992;        // 2,097,152  (512 x 4096)
  float* WT   = ws + 6406144;        // 8,388,608  (4096 x 2 x 1024)
  // total workspace ~59.2 MB

  tables_kernel<<<192, 256, 0, stream>>>(Bw, Bu, CasT);
  wtrans_kernel<<<32768, 256, 0, stream>>>(W, WT);
  fwd_kernel<<<512, 256, 0, stream>>>(x, Bu, EOm);
  mix_kernel<<<4096, 64, 0, stream>>>(EOm, WT, SPEC);
  inv_kernel<<<512, 256, 0, stream>>>(SPEC, CasT, Bw, bias, out);
}